// OrthogonalScanAttention_73452530696273
// MI455X (gfx1250) — compile-verified
//
#include <hip/hip_runtime.h>
#include <hip/hip_bf16.h>

typedef float v2f __attribute__((ext_vector_type(2)));
typedef float v4f __attribute__((ext_vector_type(4)));
typedef float v8f __attribute__((ext_vector_type(8)));
typedef int   gv4i __attribute__((vector_size(4 * sizeof(int))));

#define D_MODEL 1024
#define Hh      16
#define Nn      4
#define NSKEW   6
#define Bb      4
#define Tt      4096
#define M_ROWS  (Bb * Tt)      // 16384
#define NC1     160            // 96 skew cols + 64 v cols
#define CHUNK   64
#define NCH     (Tt / CHUNK)   // 64

#define AS1 __attribute__((address_space(1)))
#define AS3 __attribute__((address_space(3)))

// ---------------------------------------------------------------------------
// fp32 WMMA: D(16x16,f32) = A(16x4,f32) * B(4x16,f32) + C
// A frag: a[0]=A[m][2*half], a[1]=A[m][2*half+1]   (m=lane&15, half=lane>>4)
// B frag: b[0]=B[2*half][n], b[1]=B[2*half+1][n]   (n=lane&15)
// C/D:    c[r] = C[r + 8*half][n]
// ---------------------------------------------------------------------------
static __device__ inline v8f wmma_f32_16x16x4(v2f a, v2f b, v8f c) {
  return __builtin_amdgcn_wmma_f32_16x16x4_f32(false, a, false, b, (short)0, c,
                                               false, false);
}

// ---------------------------------------------------------------------------
// Async global -> LDS copy of one 16B chunk (per-lane), tracked by ASYNCcnt.
// ---------------------------------------------------------------------------
static __device__ inline void async_copy_b128(const float* g, float* l) {
#if __has_builtin(__builtin_amdgcn_global_load_async_to_lds_b128)
  __builtin_amdgcn_global_load_async_to_lds_b128(
      (AS1 gv4i*)(unsigned long long)g,
      (AS3 gv4i*)(unsigned)(unsigned long long)l, 0, 0);
#else
  unsigned lds = (unsigned)(unsigned long long)l;
  unsigned long long ga = (unsigned long long)g;
  asm volatile("global_load_async_to_lds_b128 %0, %1, off"
               :: "v"(lds), "v"(ga) : "memory");
#endif
}

static __device__ inline void wait_async0() {
#if __has_builtin(__builtin_amdgcn_s_wait_asynccnt)
  __builtin_amdgcn_s_wait_asynccnt(0);
#else
  asm volatile("s_wait_asynccnt 0x0" ::: "memory");
#endif
}

// ---------------------------------------------------------------------------
// Weight repack kernels: fragment-order pairs so one aligned load = one B frag.
// Wp  : [p<512][m<16][ct<10] pairs  (p = K/2); source [W_skew | W_v]
// Wop : [cg<16][p<32][m<16][ct<4] pairs;      source W_o (64 x 1024)
// ---------------------------------------------------------------------------
__global__ void k_pack_in(const float* __restrict__ Wsk,
                          const float* __restrict__ Wv,
                          float* __restrict__ Wp) {
  const int id = blockIdx.x * blockDim.x + threadIdx.x;
  if (id >= 512 * NC1) return;
  const int p = id / NC1, col = id % NC1;
  const int ct = col / 16, m = col % 16;
  float lo, hi;
  if (col < 96) { lo = Wsk[(size_t)(2 * p) * 96 + col];      hi = Wsk[(size_t)(2 * p + 1) * 96 + col]; }
  else          { lo = Wv[(size_t)(2 * p) * 64 + col - 96];  hi = Wv[(size_t)(2 * p + 1) * 64 + col - 96]; }
  float* d = Wp + ((size_t)(p * 16 + m) * 10 + ct) * 2;
  d[0] = lo; d[1] = hi;
}

__global__ void k_pack_out(const float* __restrict__ Wo, float* __restrict__ Wop) {
  const int id = blockIdx.x * blockDim.x + threadIdx.x;
  if (id >= 32 * D_MODEL) return;
  const int p = id / D_MODEL, col = id % D_MODEL;
  const int cg = col >> 6, w = col & 63, ct = w >> 4, m = w & 15;
  float* d = Wop + (size_t)cg * 4096 + ((size_t)(p * 16 + m) * 4 + ct) * 2;
  d[0] = Wo[(size_t)(2 * p) * D_MODEL + col];
  d[1] = Wo[(size_t)(2 * p + 1) * D_MODEL + col];
}

// ---------------------------------------------------------------------------
// Kernel 1: y = x @ [W_skew | W_v]   (16384 x 1024) * (1024 x 160), WMMA f32.
// 4 waves/block share B panels staged in LDS by async copies, double-buffered.
// ---------------------------------------------------------------------------
#define KB 64                      // K rows per LDS stage
#define PAIRS_KB (KB / 2)          // 32
#define STAGE_FLOATS (PAIRS_KB * NC1 * 2)   // 10240 floats = 40 KB

__global__ __launch_bounds__(128) void k_gemm_in(const float* __restrict__ x,
                                                 const float* __restrict__ Wp,
                                                 float* __restrict__ y) {
  __shared__ float sb[2][STAGE_FLOATS];
  const int wave = threadIdx.x >> 5;
  const int lane = threadIdx.x & 31;
  const int rowTile = blockIdx.x * 4 + wave;
  const int row0 = rowTile * 16;
  const int m = lane & 15;
  const int half = lane >> 4;

  v8f acc[10];
#pragma unroll
  for (int i = 0; i < 10; ++i)
#pragma unroll
    for (int j = 0; j < 8; ++j) acc[i][j] = 0.0f;

  const float* arow = x + (size_t)(row0 + m) * D_MODEL + 2 * half;

  // prologue: stage K-block 0 into buffer 0
#pragma unroll
  for (int j = 0; j < 20; ++j) {
    const int c16 = threadIdx.x + j * 128;          // 2560 chunks of 16B
    async_copy_b128(Wp + c16 * 4, &sb[0][c16 * 4]);
  }

  for (int kb = 0; kb < D_MODEL / KB; ++kb) {
    wait_async0();
    __syncthreads();                // stage(kb) visible; reads of other buf done
    if (kb + 1 < D_MODEL / KB) {
      const float* src = Wp + (size_t)(kb + 1) * STAGE_FLOATS;
      float* dst = &sb[(kb + 1) & 1][0];
#pragma unroll
      for (int j = 0; j < 20; ++j) {
        const int c16 = threadIdx.x + j * 128;
        async_copy_b128(src + c16 * 4, dst + c16 * 4);
      }
    }
    const float* cur = &sb[kb & 1][0];
#pragma unroll 4
    for (int k4 = 0; k4 < KB / 4; ++k4) {
      const int kg = kb * KB + k4 * 4;
      if ((kg & 255) == 0) __builtin_prefetch(arow + kg + 256, 0, 3);
      v2f a = *(const v2f*)(arow + kg);
      const int pl = k4 * 2 + half;                 // pair row in this block
      const float* base = cur + pl * 320 + m * 20;  // [pl][m][ct] pairs
#pragma unroll
      for (int ct2 = 0; ct2 < 5; ++ct2) {
        v4f f = *(const v4f*)(base + ct2 * 4);      // two B frags (2 column tiles)
        v2f b0; b0.x = f.x; b0.y = f.y;
        v2f b1; b1.x = f.z; b1.y = f.w;
        acc[2 * ct2 + 0] = wmma_f32_16x16x4(a, b0, acc[2 * ct2 + 0]);
        acc[2 * ct2 + 1] = wmma_f32_16x16x4(a, b1, acc[2 * ct2 + 1]);
      }
    }
    __syncthreads();                // done reading cur before it is re-staged
  }

#pragma unroll
  for (int ct = 0; ct < 10; ++ct)
#pragma unroll
    for (int r = 0; r < 8; ++r) {
      const int row = row0 + r + 8 * half;
      y[(size_t)row * NC1 + ct * 16 + m] = acc[ct][r];
    }
}

// ---------------------------------------------------------------------------
// 4x4 helpers
// ---------------------------------------------------------------------------
static __device__ inline void mm4(const float* X, const float* Y, float* Z) {
#pragma unroll
  for (int i = 0; i < 4; ++i)
#pragma unroll
    for (int j = 0; j < 4; ++j) {
      float s = 0.0f;
#pragma unroll
      for (int k = 0; k < 4; ++k) s = fmaf(X[i * 4 + k], Y[k * 4 + j], s);
      Z[i * 4 + j] = s;
    }
}

// ---------------------------------------------------------------------------
// Kernel 2: O[b,t,h] = expm(skew(b,t,h)), scaling-and-squaring + Horner Taylor.
// ---------------------------------------------------------------------------
__global__ void k_expm(const float* __restrict__ y, float* __restrict__ O) {
  const int idx = blockIdx.x * blockDim.x + threadIdx.x;  // (b*T+t)*H + h
  if (idx >= M_ROWS * Hh) return;
  const int h = idx % Hh;
  const int bt = idx / Hh;
  const float* s = y + (size_t)bt * NC1 + h * NSKEW;
  const float s0 = s[0], s1 = s[1], s2 = s[2], s3 = s[3], s4 = s[4], s5 = s[5];
  float A[16] = { 0.f,  s0,  s1,  s2,
                 -s0, 0.f,  s3,  s4,
                 -s1, -s3, 0.f,  s5,
                 -s2, -s4, -s5, 0.f};
  float th = fmaxf(fmaxf(fmaxf(fabsf(s0), fabsf(s1)), fmaxf(fabsf(s2), fabsf(s3))),
                   fmaxf(fabsf(s4), fabsf(s5)));
  int sc = 0;
  float tt = th;
  while (tt > 0.25f && sc < 24) { tt *= 0.5f; ++sc; }
  const float scale = exp2f((float)(-sc));
#pragma unroll
  for (int i = 0; i < 16; ++i) A[i] *= scale;

  float R[16], Tm[16];
#pragma unroll
  for (int i = 0; i < 16; ++i) R[i] = (i % 5 == 0) ? 1.0f : 0.0f;
  for (int k = 9; k >= 1; --k) {
    mm4(A, R, Tm);
    const float inv = 1.0f / (float)k;
#pragma unroll
    for (int i = 0; i < 16; ++i) R[i] = ((i % 5 == 0) ? 1.0f : 0.0f) + Tm[i] * inv;
  }
  for (int i = 0; i < sc; ++i) {
    mm4(R, R, Tm);
#pragma unroll
    for (int j = 0; j < 16; ++j) R[j] = Tm[j];
  }
  float* o = O + (size_t)idx * 16;
#pragma unroll
  for (int i = 0; i < 16; ++i) o[i] = R[i];
}

// ---------------------------------------------------------------------------
// Kernel 3: chunk-local left products  P[bh,c] = O_{c*64+63} @ ... @ O_{c*64}
// ---------------------------------------------------------------------------
__global__ void k_chunk(const float* __restrict__ O, float* __restrict__ P) {
  const int id = blockIdx.x * blockDim.x + threadIdx.x;  // (b*H+h)*NCH + c
  if (id >= Bb * Hh * NCH) return;
  const int c = id % NCH;
  const int bh = id / NCH;
  const int b = bh / Hh, h = bh % Hh;
  float M[16], Tm[16], Ot[16];
#pragma unroll
  for (int i = 0; i < 16; ++i) M[i] = (i % 5 == 0) ? 1.0f : 0.0f;
  for (int i = 0; i < CHUNK; ++i) {
    const int t = c * CHUNK + i;
    const float* o = O + ((size_t)(b * Tt + t) * Hh + h) * 16;
#pragma unroll
    for (int j = 0; j < 16; ++j) Ot[j] = o[j];
    mm4(Ot, M, Tm);
#pragma unroll
    for (int j = 0; j < 16; ++j) M[j] = Tm[j];
  }
  float* p = P + (size_t)id * 16;
#pragma unroll
  for (int i = 0; i < 16; ++i) p[i] = M[i];
}

// ---------------------------------------------------------------------------
// Kernel 4: exclusive scan of chunk totals per chain (S[bh,0] = I).
// ---------------------------------------------------------------------------
__global__ void k_scan(const float* __restrict__ P, float* __restrict__ S) {
  const int bh = blockIdx.x * blockDim.x + threadIdx.x;
  if (bh >= Bb * Hh) return;
  float run[16], Tm[16], Pm[16];
#pragma unroll
  for (int i = 0; i < 16; ++i) run[i] = (i % 5 == 0) ? 1.0f : 0.0f;
  for (int c = 0; c < NCH; ++c) {
    float* s = S + ((size_t)bh * NCH + c) * 16;
#pragma unroll
    for (int i = 0; i < 16; ++i) s[i] = run[i];
    const float* p = P + ((size_t)bh * NCH + c) * 16;
#pragma unroll
    for (int i = 0; i < 16; ++i) Pm[i] = p[i];
    mm4(Pm, run, Tm);
#pragma unroll
    for (int i = 0; i < 16; ++i) run[i] = Tm[i];
  }
}

// ---------------------------------------------------------------------------
// Kernel 5: apply — states[t] = O_t @ M (running), rotated[t] = states[t] @ v[t]
// ---------------------------------------------------------------------------
__global__ void k_apply(const float* __restrict__ O, const float* __restrict__ S,
                        const float* __restrict__ y, float* __restrict__ rot) {
  const int id = blockIdx.x * blockDim.x + threadIdx.x;  // (b*H+h)*NCH + c
  if (id >= Bb * Hh * NCH) return;
  const int c = id % NCH;
  const int bh = id / NCH;
  const int b = bh / Hh, h = bh % Hh;
  float M[16], Tm[16], Ot[16];
  const float* s = S + (size_t)id * 16;
#pragma unroll
  for (int i = 0; i < 16; ++i) M[i] = s[i];
  for (int i = 0; i < CHUNK; ++i) {
    const int t = c * CHUNK + i;
    const size_t bt = (size_t)(b * Tt + t);
    const float* o = O + (bt * Hh + h) * 16;
#pragma unroll
    for (int j = 0; j < 16; ++j) Ot[j] = o[j];
    mm4(Ot, M, Tm);
#pragma unroll
    for (int j = 0; j < 16; ++j) M[j] = Tm[j];
    const float* vv = y + bt * NC1 + 96 + h * Nn;
    const float v0 = vv[0], v1 = vv[1], v2 = vv[2], v3 = vv[3];
    float* r = rot + bt * (Hh * Nn) + h * Nn;
#pragma unroll
    for (int ii = 0; ii < 4; ++ii)
      r[ii] = fmaf(M[ii * 4 + 0], v0,
              fmaf(M[ii * 4 + 1], v1,
              fmaf(M[ii * 4 + 2], v2, M[ii * 4 + 3] * v3)));
  }
}

// ---------------------------------------------------------------------------
// Kernel 6: out = rotated(16384x64) @ W_o(64x1024), WMMA f32.
// Block = 4 row tiles x one 64-col group; B panel (16 KB) async-staged once.
// ---------------------------------------------------------------------------
__global__ __launch_bounds__(128) void k_gemm_out(const float* __restrict__ rot,
                                                  const float* __restrict__ Wop,
                                                  float* __restrict__ out) {
  __shared__ float so[4096];                       // 32 pairs x 16 m x 4 ct x 2
  const int wave = threadIdx.x >> 5;
  const int lane = threadIdx.x & 31;
  const int rowTile = (blockIdx.x >> 4) * 4 + wave;
  const int cg = blockIdx.x & 15;
  const int row0 = rowTile * 16;
  const int col0 = cg * 64;
  const int m = lane & 15;
  const int half = lane >> 4;

  {
    const float* src = Wop + (size_t)cg * 4096;
#pragma unroll
    for (int j = 0; j < 8; ++j) {
      const int c16 = threadIdx.x + j * 128;       // 1024 chunks of 16B
      async_copy_b128(src + c16 * 4, &so[c16 * 4]);
    }
  }
  wait_async0();
  __syncthreads();

  v8f acc[4];
#pragma unroll
  for (int i = 0; i < 4; ++i)
#pragma unroll
    for (int j = 0; j < 8; ++j) acc[i][j] = 0.0f;

  const float* arow = rot + (size_t)(row0 + m) * 64 + 2 * half;
#pragma unroll
  for (int k = 0; k < 64; k += 4) {
    v2f a = *(const v2f*)(arow + k);
    const int pl = (k >> 1) + half;
    const float* base = so + pl * 128 + m * 8;     // [pl][m][ct] pairs
    v4f f0 = *(const v4f*)(base);
    v4f f1 = *(const v4f*)(base + 4);
    v2f b0; b0.x = f0.x; b0.y = f0.y;
    v2f b1; b1.x = f0.z; b1.y = f0.w;
    v2f b2; b2.x = f1.x; b2.y = f1.y;
    v2f b3; b3.x = f1.z; b3.y = f1.w;
    acc[0] = wmma_f32_16x16x4(a, b0, acc[0]);
    acc[1] = wmma_f32_16x16x4(a, b1, acc[1]);
    acc[2] = wmma_f32_16x16x4(a, b2, acc[2]);
    acc[3] = wmma_f32_16x16x4(a, b3, acc[3]);
  }

#pragma unroll
  for (int ct = 0; ct < 4; ++ct)
#pragma unroll
    for (int r = 0; r < 8; ++r) {
      const int row = row0 + r + 8 * half;
      out[(size_t)row * D_MODEL + col0 + ct * 16 + m] = acc[ct][r];
    }
}

// ---------------------------------------------------------------------------
extern "C" void kernel_launch(void* const* d_in, const int* in_sizes, int n_in,
                              void* d_out, int out_size, void* d_ws, size_t ws_size,
                              hipStream_t stream) {
  const float* x   = (const float*)d_in[0];
  const float* Wsk = (const float*)d_in[1];
  const float* Wv  = (const float*)d_in[2];
  const float* Wo  = (const float*)d_in[3];
  float* out = (float*)d_out;

  char* ws = (char*)d_ws;
  size_t off = 0;
  float* y   = (float*)(ws + off); off += (size_t)M_ROWS * NC1 * sizeof(float);        // 10.5 MB
  float* O   = (float*)(ws + off); off += (size_t)M_ROWS * Hh * 16 * sizeof(float);    // 16.8 MB
  float* P   = (float*)(ws + off); off += (size_t)Bb * Hh * NCH * 16 * sizeof(float);  // 0.26 MB
  float* S   = (float*)(ws + off); off += (size_t)Bb * Hh * NCH * 16 * sizeof(float);  // 0.26 MB
  float* rot = (float*)(ws + off); off += (size_t)M_ROWS * Hh * Nn * sizeof(float);    // 4.2 MB
  float* Wp  = (float*)(ws + off); off += (size_t)512 * NC1 * 2 * sizeof(float);       // 0.64 MB
  float* Wop = (float*)(ws + off); off += (size_t)32 * D_MODEL * 2 * sizeof(float);    // 0.26 MB
  (void)ws_size; (void)in_sizes; (void)n_in; (void)out_size;

  // 0. repack weights into WMMA fragment order
  k_pack_in<<<(512 * NC1 + 255) / 256, 256, 0, stream>>>(Wsk, Wv, Wp);
  k_pack_out<<<(32 * D_MODEL + 255) / 256, 256, 0, stream>>>(Wo, Wop);
  // 1. y = x @ [W_skew | W_v]   (WMMA f32, async-LDS B panels)
  k_gemm_in<<<M_ROWS / 16 / 4, 128, 0, stream>>>(x, Wp, y);
  // 2. O = expm(skew)
  k_expm<<<(M_ROWS * Hh + 255) / 256, 256, 0, stream>>>(y, O);
  // 3. chunk-local products
  k_chunk<<<(Bb * Hh * NCH + 255) / 256, 256, 0, stream>>>(O, P);
  // 4. scan of chunk totals
  k_scan<<<1, 64, 0, stream>>>(P, S);
  // 5. apply: rotated = states @ v
  k_apply<<<(Bb * Hh * NCH + 255) / 256, 256, 0, stream>>>(O, S, y, rot);
  // 6. out = rotated @ W_o   (WMMA f32, async-LDS B panel)
  k_gemm_out<<<(M_ROWS / 16 / 4) * 16, 128, 0, stream>>>(rot, Wop, out);
}